// LocalMMDLoss_20469814132953
// MI455X (gfx1250) — compile-verified
//
#include <hip/hip_runtime.h>
#include <hip/hip_bf16.h>
#include <math.h>

typedef __attribute__((ext_vector_type(2))) float v2f;
typedef __attribute__((ext_vector_type(8))) float v8f;

#define N_ROWS 4096
#define D_FEAT 256
#define C_CLS  31
#define C_PAD  32
#define M_TOT  8192              // 2N
#define BT     128               // macro tile edge
#define NT     (M_TOT / BT)      // 64 tiles per edge
#define KC     16                // feature K-chunk
#define LDF    20                // LDS stride for feature tiles (floats)
#define LDW    36                // LDS stride for weight tiles (floats)

// workspace layout (float offsets)
#define WS_SQ      0             // 8192 row squared-norms
#define WS_COLSUM  8192          // 256 column sums of X
#define WS_BWNI    8449          // -1/bandwidth
#define WS_VALID   8450          // NaN guard flag
#define WS_SCCOEF  8452          // 32 s-side class coefs
#define WS_TCCOEF  8484          // 32 t-side class coefs
#define WS_SCNT    8516          // 32 ints: class counts in label_s
#define WS_TSUM    8548          // 32 floats: column sums of y_t
#define WS_INT_T   8580          // 32 ints: class present in argmax(y_t)
#define WS_W       8704          // 8192 x 32 unified weight factor

__device__ __forceinline__ const float* xrow(const float* fs, const float* ft, int row) {
    return (row < N_ROWS) ? fs + (size_t)row * D_FEAT
                          : ft + (size_t)(row - N_ROWS) * D_FEAT;
}

// ---------------- kernel 0: zero accumulators (ws is poisoned 0xAA) --------
__global__ void zero_small(float* ws, float* out) {
    int t = threadIdx.x;
    ws[WS_COLSUM + t] = 0.f;
    if (t < 32) {
        ws[WS_TSUM + t] = 0.f;
        ((int*)ws)[WS_SCNT + t]  = 0;
        ((int*)ws)[WS_INT_T + t] = 0;
    }
    if (t == 0) out[0] = 0.f;
}

// ---------------- kernel 1: per-row squared norms --------------------------
__global__ __launch_bounds__(256) void row_sq(const float* fs, const float* ft, float* ws) {
    int row = blockIdx.x;
    float v = xrow(fs, ft, row)[threadIdx.x];
    float p = v * v;
    for (int o = 16; o > 0; o >>= 1) p += __shfl_xor(p, o, 32);
    __shared__ float red[8];
    if ((threadIdx.x & 31) == 0) red[threadIdx.x >> 5] = p;
    __syncthreads();
    if (threadIdx.x == 0) {
        float s = 0.f;
        for (int i = 0; i < 8; i++) s += red[i];
        ws[WS_SQ + row] = s;
    }
}

// ---------------- kernel 2: column sums of X (for closed-form bandwidth) ---
__global__ __launch_bounds__(256) void colsum_k(const float* fs, const float* ft, float* ws) {
    int d = threadIdx.x;
    int r0 = blockIdx.x * 256;
    float s = 0.f;
    for (int r = 0; r < 256; r++) s += xrow(fs, ft, r0 + r)[d];
    atomicAdd(&ws[WS_COLSUM + d], s);
}

// ---------------- kernel 3: class statistics -------------------------------
__global__ __launch_bounds__(256) void class_stats(const int* label, const float* yt, float* ws) {
    __shared__ float tp[C_CLS];
    if (threadIdx.x < C_CLS) tp[threadIdx.x] = 0.f;
    __syncthreads();
    int i = blockIdx.x * blockDim.x + threadIdx.x;
    int* scnt  = (int*)ws + WS_SCNT;
    int* int_t = (int*)ws + WS_INT_T;
    if (i < N_ROWS) {
        atomicAdd(&scnt[label[i]], 1);
        const float* row = yt + (size_t)i * C_CLS;
        float best = row[0]; int bi = 0;
        for (int c = 1; c < C_CLS; c++) { float v = row[c]; if (v > best) { best = v; bi = c; } }
        atomicOr(&int_t[bi], 1);
        for (int c = 0; c < C_CLS; c++) atomicAdd(&tp[c], row[c]);   // LDS float atomics
    }
    __syncthreads();
    if (threadIdx.x < C_CLS) atomicAdd(&ws[WS_TSUM + threadIdx.x], tp[threadIdx.x]);
}

// ---------------- kernel 4: scalars (one wave) -----------------------------
__global__ void finalize_k(float* ws, float* out) {
    int c = threadIdx.x;                       // 0..31, single wave32
    int* scnt  = (int*)ws + WS_SCNT;
    int* int_t = (int*)ws + WS_INT_T;
    float scv = (c < C_CLS) ? (float)scnt[c] : 0.f;
    float tsv = (c < C_CLS) ? ws[WS_TSUM + c] : 0.f;
    float sdiv = (scv == 0.f) ? 100.f : scv;
    float tdiv = (tsv == 0.f) ? 100.f : tsv;
    int m = (c < C_CLS && scnt[c] > 0 && int_t[c] != 0) ? 1 : 0;

    float cnt = (float)m;
    for (int o = 16; o > 0; o >>= 1) cnt += __shfl_xor(cnt, o, 32);
    float denom = (cnt > 0.f) ? cnt : 1.f;
    float rsd = 1.f / sqrtf(denom);
    ws[WS_SCCOEF + c] = m ? rsd / sdiv : 0.f;
    ws[WS_TCCOEF + c] = m ? rsd / tdiv : 0.f;

    // sum of row norms and ||colsum||^2
    float ss = 0.f;
    for (int j = c; j < M_TOT; j += 32) ss += ws[WS_SQ + j];
    for (int o = 16; o > 0; o >>= 1) ss += __shfl_xor(ss, o, 32);
    float cs2 = 0.f;
    for (int j = c; j < D_FEAT; j += 32) { float v = ws[WS_COLSUM + j]; cs2 += v * v; }
    for (int o = 16; o > 0; o >>= 1) cs2 += __shfl_xor(cs2, o, 32);

    if (c == 0) {
        // sum(l2) = 2*M*sum(sq) - 2*||sum_i x_i||^2   (exact, incl. zero diagonal)
        float sum_l2 = 2.f * (float)M_TOT * ss - 2.f * cs2;
        float bw = sum_l2 / (float)((long long)M_TOT * M_TOT - M_TOT);
        float bwni = -1.f / bw;
        ws[WS_BWNI]  = bwni;
        ws[WS_VALID] = (bwni * 0.f == 0.f) ? 1.f : 0.f;  // finite check (NaN guard)
        out[0] = 0.f;
    }
}

// ---------------- kernel 5: unified weight factor W (8192 x 32) ------------
// s-rows: +onehot(label)*mask/(s_cnt*sqrt(denom));  t-rows: -y_t*mask/(t_sum*sqrt(denom))
// sign fold makes cross-side products negative => implements the -2*w_st*st term.
__global__ __launch_bounds__(256) void build_w(const int* label, const float* yt, float* ws) {
    int idx = blockIdx.x * 256 + threadIdx.x;   // < 8192*32
    int i = idx >> 5, c = idx & 31;
    float v = 0.f;
    if (c < C_CLS) {
        if (i < N_ROWS) v = (label[i] == c) ? ws[WS_SCCOEF + c] : 0.f;
        else            v = -yt[(size_t)(i - N_ROWS) * C_CLS + c] * ws[WS_TCCOEF + c];
    }
    ws[WS_W + idx] = v;
}

// ---------------- kernel 6: fused dual-Gram WMMA + exp + weighted reduce ---
__global__ __launch_bounds__(256) void mmd_main(const float* fs, const float* ft,
                                               const float* ws, float* out) {
    __shared__ float Af[BT * LDF];
    __shared__ float Bf[BT * LDF];
    __shared__ float Aw[BT * LDW];
    __shared__ float Bw[BT * LDW];
    __shared__ float sqA[BT], sqB[BT];
    __shared__ float red[8];

    // upper-triangular tile-pair decode
    int b = blockIdx.x, ti = 0, rl = NT;
    while (b >= rl) { b -= rl; ti++; rl--; }
    int tj = ti + b;
    const int rowA0 = ti * BT, rowB0 = tj * BT;
    const int tid = threadIdx.x;

    if (tid < BT) sqA[tid] = ws[WS_SQ + rowA0 + tid];
    else          sqB[tid - BT] = ws[WS_SQ + rowB0 + (tid - BT)];

    // stage weight tiles (128x32 each side)
    const float* Wm = ws + WS_W;
#pragma unroll
    for (int q = 0; q < 4; q++) {
        int j = tid + 256 * q;           // float4 id 0..1023
        int row = j >> 3;                // 8 float4 per 32-float row
        int c4 = (j & 7) * 4;
        float4 va = *reinterpret_cast<const float4*>(&Wm[(size_t)(rowA0 + row) * C_PAD + c4]);
        float4 vb = *reinterpret_cast<const float4*>(&Wm[(size_t)(rowB0 + row) * C_PAD + c4]);
        *reinterpret_cast<float4*>(&Aw[row * LDW + c4]) = va;
        *reinterpret_cast<float4*>(&Bw[row * LDW + c4]) = vb;
    }
    __syncthreads();

    const int lane = tid & 31;
    const int l16  = lane & 15;
    const int half = lane >> 4;          // A/B operand: K-pair select per ISA layout
    const int w    = tid >> 5;           // wave 0..7
    const int m0   = (w >> 2) * 64;      // 2 wave-rows
    const int n0   = (w & 3) * 32;       // 4 wave-cols

    v8f wAcc[4][2], gAcc[4][2];
    const v8f vzero = {0.f, 0.f, 0.f, 0.f, 0.f, 0.f, 0.f, 0.f};
#pragma unroll
    for (int mi = 0; mi < 4; mi++)
#pragma unroll
        for (int ni = 0; ni < 2; ni++) { wAcc[mi][ni] = vzero; gAcc[mi][ni] = vzero; }

    // ---- weight Gram: K = 32 (8 x f32 WMMA k-steps) ----
#pragma unroll
    for (int ks = 0; ks < 8; ks++) {
        const int kk = ks * 4 + 2 * half;
        v2f a[4], bb[2];
#pragma unroll
        for (int mi = 0; mi < 4; mi++)
            a[mi] = *reinterpret_cast<const v2f*>(&Aw[(m0 + mi * 16 + l16) * LDW + kk]);
#pragma unroll
        for (int ni = 0; ni < 2; ni++)
            bb[ni] = *reinterpret_cast<const v2f*>(&Bw[(n0 + ni * 16 + l16) * LDW + kk]);
#pragma unroll
        for (int mi = 0; mi < 4; mi++)
#pragma unroll
            for (int ni = 0; ni < 2; ni++)
                wAcc[mi][ni] = __builtin_amdgcn_wmma_f32_16x16x4_f32(
                    false, a[mi], false, bb[ni], (short)0, wAcc[mi][ni], false, false);
    }

    // ---- feature Gram: K = 256 in 16 LDS chunks of 16 ----
    for (int kc = 0; kc < D_FEAT / KC; kc++) {
        __syncthreads();
#pragma unroll
        for (int q = 0; q < 2; q++) {
            int j = tid + 256 * q;       // float4 id 0..511 per side
            int row = j >> 2;            // 4 float4 per 16-float row
            int c4 = (j & 3) * 4;
            int k = kc * KC + c4;
            const float* pa = xrow(fs, ft, rowA0 + row);
            const float* pb = xrow(fs, ft, rowB0 + row);
            *reinterpret_cast<float4*>(&Af[row * LDF + c4]) =
                *reinterpret_cast<const float4*>(&pa[k]);
            *reinterpret_cast<float4*>(&Bf[row * LDF + c4]) =
                *reinterpret_cast<const float4*>(&pb[k]);
        }
        __syncthreads();
#pragma unroll
        for (int ks = 0; ks < KC / 4; ks++) {
            const int kk = ks * 4 + 2 * half;
            v2f a[4], bb[2];
#pragma unroll
            for (int mi = 0; mi < 4; mi++)
                a[mi] = *reinterpret_cast<const v2f*>(&Af[(m0 + mi * 16 + l16) * LDF + kk]);
#pragma unroll
            for (int ni = 0; ni < 2; ni++)
                bb[ni] = *reinterpret_cast<const v2f*>(&Bf[(n0 + ni * 16 + l16) * LDF + kk]);
#pragma unroll
            for (int mi = 0; mi < 4; mi++)
#pragma unroll
                for (int ni = 0; ni < 2; ni++)
                    gAcc[mi][ni] = __builtin_amdgcn_wmma_f32_16x16x4_f32(
                        false, a[mi], false, bb[ni], (short)0, gAcc[mi][ni], false, false);
        }
    }
    __syncthreads();

    // ---- combine: loss += W_i.W_j * exp(-l2/bw) ----
    const float bwni = ws[WS_BWNI];
    float part = 0.f;
#pragma unroll
    for (int mi = 0; mi < 4; mi++)
#pragma unroll
        for (int ni = 0; ni < 2; ni++)
#pragma unroll
            for (int r = 0; r < 8; r++) {
                int il = m0 + mi * 16 + r + 8 * half;   // C/D layout: vgpr r -> M=r (+8 for hi half)
                int jl = n0 + ni * 16 + l16;            // N = lane 0..15
                float l2 = sqA[il] + sqB[jl] - 2.f * gAcc[mi][ni][r];
                part += wAcc[mi][ni][r] * __expf(l2 * bwni);
            }

    for (int o = 16; o > 0; o >>= 1) part += __shfl_xor(part, o, 32);
    if (lane == 0) red[w] = part;
    __syncthreads();
    if (tid == 0) {
        float s = 0.f;
        for (int i = 0; i < 8; i++) s += red[i];
        float fac = (ti == tj) ? 1.f : 2.f;             // symmetry: off-diag tiles counted twice
        atomicAdd(out, s * fac * ws[WS_VALID]);
    }
}

extern "C" void kernel_launch(void* const* d_in, const int* in_sizes, int n_in,
                              void* d_out, int out_size, void* d_ws, size_t ws_size,
                              hipStream_t stream) {
    const float* fs    = (const float*)d_in[0];
    const float* ft    = (const float*)d_in[1];
    const int*   label = (const int*)d_in[2];
    const float* yt    = (const float*)d_in[3];
    float* ws  = (float*)d_ws;
    float* out = (float*)d_out;
    (void)in_sizes; (void)n_in; (void)out_size; (void)ws_size;

    zero_small  <<<1, 256, 0, stream>>>(ws, out);
    row_sq      <<<M_TOT, 256, 0, stream>>>(fs, ft, ws);
    colsum_k    <<<M_TOT / 256, 256, 0, stream>>>(fs, ft, ws);
    class_stats <<<N_ROWS / 256, 256, 0, stream>>>(label, yt, ws);
    finalize_k  <<<1, 32, 0, stream>>>(ws, out);
    build_w     <<<(M_TOT * C_PAD) / 256, 256, 0, stream>>>(label, yt, ws);
    mmd_main    <<<NT * (NT + 1) / 2, 256, 0, stream>>>(fs, ft, ws, out);
}